// AllExamplesPairwiseMSELossAndBCEWithLogitsLoss_85100482003008
// MI455X (gfx1250) — compile-verified
//
#include <hip/hip_runtime.h>
#include <hip/hip_bf16.h>
#include <math.h>

typedef __attribute__((ext_vector_type(2))) float v2f;
typedef __attribute__((ext_vector_type(8))) float v8f;

#define LOGIT_EPS      1e-7f
#define DPSI_THRESHOLD 0.05f
#define MSE_WEIGHT     10.0f
#define TILE_M         16
#define WAVES_PER_ROW  8
#define UNROLL         8

// ---------------------------------------------------------------------------
// Kernel 1: O(N) prep — per-block BCE partial sums + e[i] = pred[i]-logit(psi)
// ---------------------------------------------------------------------------
__global__ __launch_bounds__(256) void prep_kernel(const float* __restrict__ pred,
                                                   const float* __restrict__ psi,
                                                   float* __restrict__ e,
                                                   float* __restrict__ bcePartial,
                                                   int n) {
  int i = blockIdx.x * 256 + threadIdx.x;
  float x = (i < n) ? pred[i] : 0.0f;
  float y = (i < n) ? psi[i]  : 0.0f;
  float term = (i < n) ? (fmaxf(x, 0.0f) - x * y + log1pf(expf(-fabsf(x)))) : 0.0f;

  __shared__ float red[256];
  red[threadIdx.x] = term;
  __syncthreads();
#pragma unroll
  for (int s = 128; s > 0; s >>= 1) {
    if (threadIdx.x < s) red[threadIdx.x] += red[threadIdx.x + s];
    __syncthreads();
  }
  if (threadIdx.x == 0) bcePartial[blockIdx.x] = red[0];

  if (i < n) {
    float p = fminf(fmaxf(y, LOGIT_EPS), 1.0f - LOGIT_EPS);
    e[i] = x - (logf(p) - log1pf(-p));
  }
}

// ---------------------------------------------------------------------------
// Kernel 2: O(N^2) pairwise via WMMA f32 16x16x4, fully branchless hot loop.
// One wave per block. wid -> rowTile = wid/8 (16 rows), jslot = wid%8
// (jchunk-wide j range). A = mask tile (16x4, exact 0/1 f32); B column n
// holds [1, e_j, e_j^2, 0...] selected by lane-constant weights; C columns
// 0/1/2 accumulate R_i, S_i, T_i.
// ---------------------------------------------------------------------------
__global__ __launch_bounds__(32) void pair_kernel(const float* __restrict__ psi,
                                                  const float* __restrict__ e,
                                                  float* __restrict__ pSumSq,
                                                  float* __restrict__ pNValid,
                                                  int jchunk) {
  const int wid     = blockIdx.x;
  const int rowTile = wid / WAVES_PER_ROW;
  const int jslot   = wid % WAVES_PER_ROW;
  const int rowBase = rowTile * TILE_M;
  const int j0base  = jslot * jchunk;
  const int lane    = (int)threadIdx.x;          // 0..31
  const int col     = lane & 15;                 // A row M / B,C column N
  const int kb      = (lane >> 4) << 1;          // this lane's K pair: {0,1} or {2,3}

  const float psiRow = psi[rowBase + col];

  // lane-constant statistic-selector weights: B[k,n] = w0 + e*w1 + e^2*w2
  const float w0 = (col == 0) ? 1.0f : 0.0f;
  const float w1 = (col == 1) ? 1.0f : 0.0f;
  const float w2 = (col == 2) ? 1.0f : 0.0f;

  const float* __restrict__ psiP = psi + j0base + kb;
  const float* __restrict__ eP   = e   + j0base + kb;

  v8f c = {0.0f, 0.0f, 0.0f, 0.0f, 0.0f, 0.0f, 0.0f, 0.0f};

  for (int j0 = 0; j0 < jchunk; j0 += 4 * UNROLL) {
#pragma unroll
    for (int u = 0; u < UNROLL; ++u) {
      const int o = j0 + 4 * u;
      float2 pj = *(const float2*)(psiP + o);
      float2 ej = *(const float2*)(eP + o);

      v2f a, b;
      // mask: exact 0/1 in f32 (sub + abs-compare + cndmask, no branches)
      a.x = (fabsf(psiRow - pj.x) >= DPSI_THRESHOLD) ? 1.0f : 0.0f;
      a.y = (fabsf(psiRow - pj.y) >= DPSI_THRESHOLD) ? 1.0f : 0.0f;
      // B value: 1 / e / e^2 / 0 by column, two FMAs per element, exact
      b.x = __builtin_fmaf(ej.x, __builtin_fmaf(ej.x, w2, w1), w0);
      b.y = __builtin_fmaf(ej.y, __builtin_fmaf(ej.y, w2, w1), w0);

      c = __builtin_amdgcn_wmma_f32_16x16x4_f32(false, a, false, b,
                                                (short)0, c, false, false);
    }
  }

  // Branchless epilogue: C VGPR v holds row rowBase+v (lanes 0-15) or
  // rowBase+v+8 (lanes 16-31); column = lane&15 picks R/S/T.
  // Contribution = x * (w0*e_i^2 - 2*w1*e_i + w2), n_valid += w0*x.
  float sumsq = 0.0f, nval = 0.0f;
  const int hiOff = (lane >> 4) * 8;
#pragma unroll
  for (int v = 0; v < 8; ++v) {
    float x    = c[v];
    float er   = e[rowBase + v + hiOff];
    float coef = __builtin_fmaf(w0, er * er, __builtin_fmaf(-2.0f * w1, er, w2));
    sumsq = __builtin_fmaf(x, coef, sumsq);
    nval  = __builtin_fmaf(w0, x, nval);
  }

  // wave32 butterfly reduction
#pragma unroll
  for (int off = 16; off > 0; off >>= 1) {
    sumsq += __shfl_xor(sumsq, off, 32);
    nval  += __shfl_xor(nval,  off, 32);
  }
  if (lane == 0) {
    pSumSq[wid]  = sumsq;
    pNValid[wid] = nval;
  }
}

// ---------------------------------------------------------------------------
// Kernel 3: deterministic final combine (double accumulation; n_valid ~ 6.7e7
// exceeds f32 exact-integer range).
// ---------------------------------------------------------------------------
__global__ __launch_bounds__(256) void final_kernel(const float* __restrict__ bcePartial, int nBce,
                                                    const float* __restrict__ pSumSq,
                                                    const float* __restrict__ pNValid, int nPair,
                                                    const int* __restrict__ flag,
                                                    float* __restrict__ out, int n) {
  __shared__ double s1[256];
  __shared__ double s2[256];
  double a = 0.0, b = 0.0;
  for (int i = threadIdx.x; i < nPair; i += 256) {
    a += (double)pSumSq[i];
    b += (double)pNValid[i];
  }
  s1[threadIdx.x] = a;
  s2[threadIdx.x] = b;
  __syncthreads();
  for (int s = 128; s > 0; s >>= 1) {
    if (threadIdx.x < s) {
      s1[threadIdx.x] += s1[threadIdx.x + s];
      s2[threadIdx.x] += s2[threadIdx.x + s];
    }
    __syncthreads();
  }
  if (threadIdx.x == 0) {
    double bce = 0.0;
    for (int i = 0; i < nBce; ++i) bce += (double)bcePartial[i];
    float loss = (float)(bce / (double)n);
    double nv  = s2[0];
    float res  = loss;
    if ((*flag) == 0 && nv > 0.0) {
      float mse = (float)(s1[0] / (nv < 1.0 ? 1.0 : nv));
      res = loss + MSE_WEIGHT * mse;
    }
    out[0] = res;
  }
}

// ---------------------------------------------------------------------------
extern "C" void kernel_launch(void* const* d_in, const int* in_sizes, int n_in,
                              void* d_out, int out_size, void* d_ws, size_t ws_size,
                              hipStream_t stream) {
  const float* pred = (const float*)d_in[0];
  const float* psi  = (const float*)d_in[1];
  const int*   flag = (const int*)d_in[2];
  float* out = (float*)d_out;

  const int n           = in_sizes[0];               // 8192
  const int nBlocksPrep = (n + 255) / 256;           // 32
  const int nRowTiles   = n / TILE_M;                // 512
  const int nPair       = nRowTiles * WAVES_PER_ROW; // 4096 single-wave blocks
  const int jchunk      = n / WAVES_PER_ROW;         // 1024

  // workspace layout (floats): e[n] | bcePartial | pSumSq[nPair] | pNValid[nPair]
  float* ws         = (float*)d_ws;
  float* e          = ws;
  float* bcePartial = ws + n;
  float* pSumSq     = bcePartial + ((nBlocksPrep + 63) & ~63);
  float* pNValid    = pSumSq + nPair;

  prep_kernel<<<nBlocksPrep, 256, 0, stream>>>(pred, psi, e, bcePartial, n);
  pair_kernel<<<nPair, 32, 0, stream>>>(psi, e, pSumSq, pNValid, jchunk);
  final_kernel<<<1, 256, 0, stream>>>(bcePartial, nBlocksPrep, pSumSq, pNValid,
                                      nPair, flag, out, n);
}